// NodeMPNN_32890859553196
// MI455X (gfx1250) — compile-verified
//
#include <hip/hip_runtime.h>
#include <math.h>

#define DIM_H 128
#define LN_EPS 1e-5f

typedef __attribute__((ext_vector_type(2))) float v2f;
typedef __attribute__((ext_vector_type(8))) float v8f;

__device__ __forceinline__ v8f wmma4(v2f a, v2f b, v8f c) {
    // D = A(16x4 f32) x B(4x16 f32) + C(16x16 f32)
    return __builtin_amdgcn_wmma_f32_16x16x4_f32(
        /*neg_a=*/false, a, /*neg_b=*/false, b,
        /*c_mod=*/(short)0, c, /*reuse_a=*/false, /*reuse_b=*/false);
}

// ---------------------------------------------------------------- zero
__global__ void zero_kernel(float* __restrict__ p, long n) {
    long i = (long)blockIdx.x * blockDim.x + threadIdx.x;
    long stride = (long)gridDim.x * blockDim.x;
    for (; i < n; i += stride) p[i] = 0.0f;
}

// ---------------------------------------------------------------- column sums of nodes (for mean)
__global__ void colsum_kernel(const float* __restrict__ nodes,
                              float* __restrict__ colsum, int n_nodes) {
    int col = threadIdx.x;  // blockDim.x == 128
    float s = 0.0f;
    for (int r = blockIdx.x; r < n_nodes; r += gridDim.x)
        s += nodes[(long)r * DIM_H + col];
    atomicAdd(&colsum[col], s);
}

// ---------------------------------------------------------------- P = nodes @ W_msg^T + b_msg
__global__ void __launch_bounds__(128)
proj_kernel(const float* __restrict__ nodes, const float* __restrict__ Wmsg,
            const float* __restrict__ bmsg, float* __restrict__ P,
            int n_nodes, int n_tiles) {
    int wave = threadIdx.x >> 5;
    int lane = threadIdx.x & 31;
    int tile = blockIdx.x * 4 + wave;
    if (tile >= n_tiles) return;

    int row0  = tile * 16;
    int m     = lane & 15;          // M index (also N index for B frags)
    int half  = lane >> 4;          // 0: K+{0,1}, 1: K+{2,3}
    int khalf = half << 1;
    int arow  = row0 + m; if (arow >= n_nodes) arow = n_nodes - 1;

    // Preload 16x128 A tile as 32 k-step fragments (64 VGPRs)
    v2f a[32];
#pragma unroll
    for (int kb = 0; kb < 32; ++kb)
        a[kb] = *(const v2f*)(nodes + (long)arow * DIM_H + kb * 4 + khalf);

    for (int nc = 0; nc < 8; ++nc) {
        int n0 = nc * 16;
        v8f acc = {};
#pragma unroll
        for (int kb = 0; kb < 32; ++kb) {
            v2f b = *(const v2f*)(Wmsg + (long)(n0 + m) * DIM_H + kb * 4 + khalf);
            acc = wmma4(a[kb], b, acc);
        }
        int   col = n0 + m;
        float bb  = bmsg[col];
#pragma unroll
        for (int e = 0; e < 8; ++e) {
            int row = row0 + e + 8 * half;   // D layout: vgpr e -> rows e / e+8
            if (row < n_nodes) P[(long)row * DIM_H + col] = acc[e] + bb;
        }
    }
}

// ---------------------------------------------------------------- messages[dst] += P[src]
__global__ void scatter_kernel(const float* __restrict__ P,
                               const int* __restrict__ esrc,
                               const int* __restrict__ edst,
                               float* __restrict__ msgs, int n_edges) {
    long total  = (long)n_edges * 32;
    long idx    = (long)blockIdx.x * blockDim.x + threadIdx.x;
    long stride = (long)gridDim.x * blockDim.x;
    for (; idx < total; idx += stride) {
        int e = (int)(idx >> 5);
        int j = (int)(idx & 31);          // 4 floats per lane; 32 lanes = full row
        int s = esrc[e], d = edst[e];
        float4 p = *(const float4*)(P + (long)s * DIM_H + j * 4);
        float* dp = msgs + (long)d * DIM_H + j * 4;
        atomicAdd(dp + 0, p.x);
        atomicAdd(dp + 1, p.y);
        atomicAdd(dp + 2, p.z);
        atomicAdd(dp + 3, p.w);
    }
}

// ---------------------------------------------------------------- fused GRU + LayerNorm + residual
__global__ void __launch_bounds__(128)
gru_kernel(const float* __restrict__ nodes, const float* __restrict__ msgs,
           const float* __restrict__ colsum,
           const float* __restrict__ w_ih, const float* __restrict__ w_hh,
           const float* __restrict__ b_ih, const float* __restrict__ b_hh,
           const float* __restrict__ gamma, const float* __restrict__ beta,
           float* __restrict__ out, int n_nodes, int n_tiles, float invN) {
    __shared__ float hbuf[4 * 16 * DIM_H];      // 32 KB: h_next tile per wave
    int wave = threadIdx.x >> 5;
    int lane = threadIdx.x & 31;
    int tile = blockIdx.x * 4 + wave;
    float* hl = hbuf + wave * 16 * DIM_H;

    int  m     = lane & 15;
    int  half  = lane >> 4;
    int  khalf = half << 1;
    bool active = (tile < n_tiles);
    int  row0  = active ? tile * 16 : 0;
    int  arow  = row0 + m; if (arow >= n_nodes) arow = n_nodes - 1;

    if (active) {
        // Preload A fragments: msg = messages + colmean, and nodes (h_prev)
        v2f amsg[32], anod[32];
#pragma unroll
        for (int kb = 0; kb < 32; ++kb) {
            int k = kb * 4 + khalf;
            v2f pm = *(const v2f*)(msgs  + (long)arow * DIM_H + k);
            v2f pn = *(const v2f*)(nodes + (long)arow * DIM_H + k);
            pm.x += colsum[k] * invN;
            pm.y += colsum[k + 1] * invN;
            amsg[kb] = pm;
            anod[kb] = pn;
        }

        for (int nc = 0; nc < 8; ++nc) {
            int n0 = nc * 16;
            v8f ir = {}, iz = {}, inn = {}, hr = {}, hz = {}, hn = {};
#pragma unroll
            for (int kb = 0; kb < 32; ++kb) {
                int k = kb * 4 + khalf;
                // torch gate layout: rows [0,128)=r, [128,256)=z, [256,384)=n
                v2f br = *(const v2f*)(w_ih + (long)(            n0 + m) * DIM_H + k);
                v2f bz = *(const v2f*)(w_ih + (long)(    DIM_H + n0 + m) * DIM_H + k);
                v2f bn = *(const v2f*)(w_ih + (long)(2 * DIM_H + n0 + m) * DIM_H + k);
                v2f cr = *(const v2f*)(w_hh + (long)(            n0 + m) * DIM_H + k);
                v2f cz = *(const v2f*)(w_hh + (long)(    DIM_H + n0 + m) * DIM_H + k);
                v2f cn = *(const v2f*)(w_hh + (long)(2 * DIM_H + n0 + m) * DIM_H + k);
                ir  = wmma4(amsg[kb], br, ir);
                iz  = wmma4(amsg[kb], bz, iz);
                inn = wmma4(amsg[kb], bn, inn);
                hr  = wmma4(anod[kb], cr, hr);
                hz  = wmma4(anod[kb], cz, hz);
                hn  = wmma4(anod[kb], cn, hn);
            }
            int   col = n0 + m;
            float bir = b_ih[col], biz = b_ih[DIM_H + col], bin = b_ih[2 * DIM_H + col];
            float bhr = b_hh[col], bhz = b_hh[DIM_H + col], bhn = b_hh[2 * DIM_H + col];
#pragma unroll
            for (int e = 0; e < 8; ++e) {
                int row = row0 + e + 8 * half;
                int rc  = row < n_nodes ? row : n_nodes - 1;
                float hprev = nodes[(long)rc * DIM_H + col];
                float r  = 1.0f / (1.0f + expf(-(ir[e] + bir + hr[e] + bhr)));
                float z  = 1.0f / (1.0f + expf(-(iz[e] + biz + hz[e] + bhz)));
                float nn = tanhf(inn[e] + bin + r * (hn[e] + bhn));
                float hnext = (1.0f - z) * nn + z * hprev;
                hl[(e + 8 * half) * DIM_H + col] = hnext;
            }
        }
    }
    __syncthreads();

    if (active) {
        // LayerNorm: 2 lanes per row (lane, lane+16), 64 cols each
        int row = lane & 15;
        int co  = half * 64;
        float s = 0.0f, s2 = 0.0f;
        for (int c = 0; c < 64; ++c) {
            float v = hl[row * DIM_H + co + c];
            s += v; s2 += v * v;
        }
        s  += __shfl_xor(s, 16, 32);
        s2 += __shfl_xor(s2, 16, 32);
        float mu   = s * (1.0f / DIM_H);
        float var  = s2 * (1.0f / DIM_H) - mu * mu;
        float rstd = rsqrtf(var + LN_EPS);
        int grow = row0 + row;
        if (grow < n_nodes) {
            for (int c = 0; c < 64; ++c) {
                int col = co + c;
                float v = hl[row * DIM_H + col];
                float o = gamma[col] * (v - mu) * rstd + beta[col]
                        + msgs[(long)grow * DIM_H + col];
                out[(long)grow * DIM_H + col] = o;
            }
        }
    }
}

// ---------------------------------------------------------------- launcher
extern "C" void kernel_launch(void* const* d_in, const int* in_sizes, int n_in,
                              void* d_out, int out_size, void* d_ws, size_t ws_size,
                              hipStream_t stream) {
    const float* nodes = (const float*)d_in[0];
    const float* W_msg = (const float*)d_in[1];
    const float* b_msg = (const float*)d_in[2];
    const float* w_ih  = (const float*)d_in[3];
    const float* w_hh  = (const float*)d_in[4];
    const float* b_ih  = (const float*)d_in[5];
    const float* b_hh  = (const float*)d_in[6];
    const float* ln_g  = (const float*)d_in[7];
    const float* ln_b  = (const float*)d_in[8];
    const int*   esrc  = (const int*)d_in[9];
    const int*   edst  = (const int*)d_in[10];

    int n_nodes = in_sizes[0] / DIM_H;
    int n_edges = in_sizes[9];
    int n_tiles = (n_nodes + 15) / 16;

    float* out    = (float*)d_out;
    float* colsum = (float*)d_ws;           // [128]
    float* msgs   = colsum + DIM_H;         // [n_nodes * 128], contiguous after colsum
    float* P      = out;                    // reuse d_out as projection scratch (dead before gru)

    // zero colsum + messages (contiguous) — must run every call (graph replay)
    long zn = (long)n_nodes * DIM_H + DIM_H;
    zero_kernel<<<1024, 256, 0, stream>>>(colsum, zn);

    colsum_kernel<<<512, DIM_H, 0, stream>>>(nodes, colsum, n_nodes);

    proj_kernel<<<(n_tiles + 3) / 4, 128, 0, stream>>>(nodes, W_msg, b_msg, P,
                                                       n_nodes, n_tiles);

    scatter_kernel<<<4096, 256, 0, stream>>>(P, esrc, edst, msgs, n_edges);

    gru_kernel<<<(n_tiles + 3) / 4, 128, 0, stream>>>(nodes, msgs, colsum,
                                                      w_ih, w_hh, b_ih, b_hh,
                                                      ln_g, ln_b, out,
                                                      n_nodes, n_tiles,
                                                      1.0f / (float)n_nodes);
}